// SAP_73323681677755
// MI455X (gfx1250) — compile-verified
//
#include <hip/hip_runtime.h>

// Problem constants from the reference: x is (64,128,64,64) f32.
#define B_ROWS 64
#define N_COLS 524288            // 128*64*64 = 2^19
#define KEEP   262144            // N * 0.5  = 2^18
#define CHUNK  2048              // elements per scan block (8 waves * 256)
#define CHUNKS_PER_ROW (N_COLS / CHUNK)        // 256
#define TOTAL_CHUNKS   (B_ROWS * CHUNKS_PER_ROW) // 16384
#define N_TOTAL (B_ROWS * N_COLS)              // 2^25 elements

typedef __attribute__((ext_vector_type(2))) float v2f;
typedef __attribute__((ext_vector_type(8))) float v8f;
typedef __attribute__((ext_vector_type(4))) unsigned int v4u;
typedef __attribute__((ext_vector_type(8))) int v8i;
typedef __attribute__((ext_vector_type(4))) int v4i;

// ---------------------------------------------------------------------------
// K1: per-chunk sums of |x|  (one block = one 2048-element chunk)
// ---------------------------------------------------------------------------
__global__ void k_chunksum(const float* __restrict__ x, float* __restrict__ partial) {
    __shared__ float red[256];
    const int tid = threadIdx.x;
    const size_t base = (size_t)blockIdx.x * CHUNK + (size_t)tid * 8;
    const float4* xp = (const float4*)(x + base);
    float4 a = xp[0], b = xp[1];
    float s = fabsf(a.x) + fabsf(a.y) + fabsf(a.z) + fabsf(a.w)
            + fabsf(b.x) + fabsf(b.y) + fabsf(b.z) + fabsf(b.w);
    red[tid] = s;
    __syncthreads();
    for (int off = 128; off > 0; off >>= 1) {
        if (tid < off) red[tid] += red[tid + off];
        __syncthreads();
    }
    if (tid == 0) partial[blockIdx.x] = red[0];
}

// ---------------------------------------------------------------------------
// K2: per-row exclusive scan of the 256 chunk sums; also emits row totals.
// ---------------------------------------------------------------------------
__global__ void k_scanpart(const float* __restrict__ partial,
                           float* __restrict__ chunkOff,
                           float* __restrict__ totals) {
    __shared__ float sc[256];
    const int tid = threadIdx.x;
    const int row = blockIdx.x;
    float v = partial[row * 256 + tid];
    float xv = v;
    sc[tid] = xv;
    __syncthreads();
    for (int off = 1; off < 256; off <<= 1) {
        float t = (tid >= off) ? sc[tid - off] : 0.0f;
        __syncthreads();
        xv += t;
        sc[tid] = xv;
        __syncthreads();
    }
    chunkOff[row * 256 + tid] = xv - v;   // exclusive prefix
    if (tid == 255) totals[row] = xv;     // full row sum
}

// ---------------------------------------------------------------------------
// K3: chunk-local inclusive scan via triangular WMMA matmuls + global offset.
// The 8 KB input tile is staged into LDS by the Tensor Data Mover (one
// tensor_load_to_lds issued by wave 0; 1-D descriptor, data_size=4B,
// tile_dim0=2048), tracked with TENSORcnt, instead of 64 b128 loads + stores.
// Each wave then scans a 16x16 tile (256 elems) from LDS:
//   R = X * U          (U upper-tri ones, K accumulated as 4 slices of K=4)
//   S = Lstrict * T + R (T = row-totals broadcast; gives inter-row offsets)
// A-layout (16x4 f32): lane l -> M = l&15, K = 2*(l>=16) + vgpr.
// B-layout (4x16 f32): lane l -> N = l&15, K = 2*(l>=16) + vgpr.
// C/D-layout: lane l, vgpr v -> (M = v + 8*(l>=16), N = l&15).
// ---------------------------------------------------------------------------
__global__ void k_scan_wmma(const float* __restrict__ x,
                            const float* __restrict__ chunkOff,
                            float* __restrict__ cdf) {
    __shared__ float sh[CHUNK];      // 8 KB tile of x (raw; abs applied at gather)
    __shared__ float tsh[8 * 16];    // per-wave row totals t[0..15]
    __shared__ float wsum[8];        // per-wave tile totals
    const int tid  = threadIdx.x;
    const int w    = tid >> 5;
    const int lane = tid & 31;
    const int c    = lane & 15;      // column (B/C/D) == A-row index m
    const int hi   = lane >> 4;
    const size_t base = (size_t)blockIdx.x * CHUNK;

    // ---- TDM: DMA the whole 2048-float tile into LDS (wave 0 issues) ----
    if (tid < 32) {
        const unsigned long long ga = (unsigned long long)(const void*)(x + base);
        const unsigned lds_off = (unsigned)(unsigned long long)(uintptr_t)&sh[0];

        v4u g0;                          // D# group 0 (128b)
        g0[0] = 1u;                                      // count=1, user D#
        g0[1] = lds_off;                                 // lds_addr
        g0[2] = (unsigned)(ga & 0xFFFFFFFFu);            // global_addr[31:0]
        g0[3] = (unsigned)((ga >> 32) & 0x01FFFFFFu)     // global_addr[56:32]
              | (2u << 30);                              // type = 2 (image)

        v8i g1;                          // D# group 1 (256b)
        g1[0] = (int)(2u << 16);         // wg_mask=0 (not in cluster), data_size=4B
        g1[1] = 0;                       // barrier addr=0; tensor_dim0[15:0]=0
        g1[2] = (int)((N_TOTAL >> 16) & 0xFFFFu)         // tensor_dim0[31:16]
              | (1 << 16);                               // tensor_dim1 = 1
        g1[3] = (int)((unsigned)CHUNK << 16);            // tile_dim0 = 2048
        g1[4] = 0;                       // tile_dim1 = 0 (unused), tile_dim2 = 0
        g1[5] = (int)N_TOTAL;            // tensor_dim0_stride[31:0]
        g1[6] = 0;                       // stride high bits / dim1_stride = 0
        g1[7] = 0;

        v4i gz = {0, 0, 0, 0};           // groups 2/3: NULL (<=2D tensor)
#if defined(__clang_major__) && (__clang_major__ >= 23)
        v8i gz8 = {0, 0, 0, 0, 0, 0, 0, 0};
        __builtin_amdgcn_tensor_load_to_lds(g0, g1, gz, gz, gz8, 0);
#else
        __builtin_amdgcn_tensor_load_to_lds(g0, g1, gz, gz, 0);
#endif
        __builtin_amdgcn_s_wait_tensorcnt(0);
    }
    __syncthreads();

    const float* tile = sh + w * 256;

    // ---- Step 1: row prefix sums  R = |X| * U ----
    v8f acc = {0.f, 0.f, 0.f, 0.f, 0.f, 0.f, 0.f, 0.f};
#pragma unroll
    for (int kk = 0; kk < 4; ++kk) {
        const int k0 = kk * 4 + hi * 2;
        v2f A;  A.x = fabsf(tile[c * 16 + k0]);
                A.y = fabsf(tile[c * 16 + k0 + 1]);
        v2f Bm; Bm.x = (k0     <= c) ? 1.0f : 0.0f;   // U[k][c] = (k <= c)
                Bm.y = (k0 + 1 <= c) ? 1.0f : 0.0f;
        acc = __builtin_amdgcn_wmma_f32_16x16x4_f32(false, A, false, Bm,
                                                    (short)0, acc, false, false);
    }

    // row totals t[m] = R[m][15] live in lanes 15 (m=0..7) and 31 (m=8..15)
    if (lane == 15) {
#pragma unroll
        for (int v = 0; v < 8; ++v) tsh[w * 16 + v] = acc[v];
    }
    if (lane == 31) {
#pragma unroll
        for (int v = 0; v < 8; ++v) tsh[w * 16 + 8 + v] = acc[v];
    }
    __syncthreads();

    // ---- Step 2: S = Lstrict * T + R ----
    v8f s = acc;
#pragma unroll
    for (int kk = 0; kk < 4; ++kk) {
        const int k0 = kk * 4 + hi * 2;
        v2f A;  A.x = (k0     < c) ? 1.0f : 0.0f;     // Lstrict[m][k] = (k < m), m == c
                A.y = (k0 + 1 < c) ? 1.0f : 0.0f;
        v2f Bm; Bm.x = tsh[w * 16 + k0];              // T[k][*] = t[k]
                Bm.y = tsh[w * 16 + k0 + 1];
        s = __builtin_amdgcn_wmma_f32_16x16x4_f32(false, A, false, Bm,
                                                  (short)0, s, false, false);
    }

    if (lane == 31) wsum[w] = s[7];  // S[15][15] = wave tile total
    __syncthreads();

    float off = chunkOff[blockIdx.x];
    for (int i = 0; i < w; ++i) off += wsum[i];

#pragma unroll
    for (int v = 0; v < 8; ++v) {
        const int e = (v + 8 * hi) * 16 + c;          // flat element in 256-tile
        cdf[base + (size_t)w * 256 + e] = s[v] + off;
    }
}

// ---------------------------------------------------------------------------
// K4: out = x / max(1 - (1-|x|/total)^keep, 1e-4)   (float4 streaming)
// ---------------------------------------------------------------------------
__global__ void k_scale(const float* __restrict__ x,
                        const float* __restrict__ totals,
                        float* __restrict__ out) {
    const size_t i4 = (size_t)blockIdx.x * blockDim.x + threadIdx.x; // 2^23 float4s
    const int row = (int)(i4 >> 17);                                  // 2^17 float4s/row
    const float inv = 1.0f / totals[row];
    const float keepf = 262144.0f;
    float4 v = ((const float4*)x)[i4];
    float4 o;
    {
        float p = fabsf(v.x) * inv;
        float sc = fmaxf(-expm1f(keepf * log1pf(-p)), 1e-4f);
        o.x = v.x / sc;
    }
    {
        float p = fabsf(v.y) * inv;
        float sc = fmaxf(-expm1f(keepf * log1pf(-p)), 1e-4f);
        o.y = v.y / sc;
    }
    {
        float p = fabsf(v.z) * inv;
        float sc = fmaxf(-expm1f(keepf * log1pf(-p)), 1e-4f);
        o.z = v.z / sc;
    }
    {
        float p = fabsf(v.w) * inv;
        float sc = fmaxf(-expm1f(keepf * log1pf(-p)), 1e-4f);
        o.w = v.w / sc;
    }
    ((float4*)out)[i4] = o;
}

// ---------------------------------------------------------------------------
// K5: JAX threefry2x32 sampling + searchsorted; zero sampled entries of out.
// jax.random.key(42) -> key = {0, 42}. counts = iota(2^24) split in halves:
// sample s<half uses out0 of threefry(key, s, s+half); s>=half uses out1.
// ---------------------------------------------------------------------------
__device__ __forceinline__ unsigned rotl32(unsigned v, int d) {
    return (v << d) | (v >> (32 - d));
}

__device__ __forceinline__ void threefry2x32(unsigned k0, unsigned k1,
                                             unsigned x0, unsigned x1,
                                             unsigned& o0, unsigned& o1) {
    const unsigned k2 = k0 ^ k1 ^ 0x1BD11BDAu;
    const int rA[4] = {13, 15, 26, 6};
    const int rB[4] = {17, 29, 16, 24};
    x0 += k0; x1 += k1;
#pragma unroll
    for (int i = 0; i < 4; ++i) { x0 += x1; x1 = rotl32(x1, rA[i]); x1 ^= x0; }
    x0 += k1; x1 += k2 + 1u;
#pragma unroll
    for (int i = 0; i < 4; ++i) { x0 += x1; x1 = rotl32(x1, rB[i]); x1 ^= x0; }
    x0 += k2; x1 += k0 + 2u;
#pragma unroll
    for (int i = 0; i < 4; ++i) { x0 += x1; x1 = rotl32(x1, rA[i]); x1 ^= x0; }
    x0 += k0; x1 += k1 + 3u;
#pragma unroll
    for (int i = 0; i < 4; ++i) { x0 += x1; x1 = rotl32(x1, rB[i]); x1 ^= x0; }
    x0 += k1; x1 += k2 + 4u;
#pragma unroll
    for (int i = 0; i < 4; ++i) { x0 += x1; x1 = rotl32(x1, rA[i]); x1 ^= x0; }
    x0 += k2; x1 += k0 + 5u;
    o0 = x0; o1 = x1;
}

__global__ void k_sample(const float* __restrict__ cdf,
                         const float* __restrict__ totals,
                         float* __restrict__ out) {
    const unsigned t = blockIdx.x * blockDim.x + threadIdx.x; // 0 .. 2^23-1
    const unsigned half = 8388608u;                            // 2^23
    unsigned o0, o1;
    threefry2x32(0u, 42u, t, t + half, o0, o1);
#pragma unroll
    for (int which = 0; which < 2; ++which) {
        const unsigned s    = which ? (t + half) : t;
        const unsigned bits = which ? o1 : o0;
        const unsigned b = s >> 18;                            // keep = 2^18
        const float u = __uint_as_float((bits >> 9) | 0x3F800000u) - 1.0f;
        const float tgt = u * totals[b];
        const float* c = cdf + ((size_t)b << 19);
        unsigned lo = 0, hiN = 1u << 19;
        while (lo < hiN) {                                     // searchsorted 'right'
            const unsigned mid = (lo + hiN) >> 1;
            if (c[mid] <= tgt) lo = mid + 1; else hiN = mid;
        }
        if (lo >= (1u << 19)) lo = (1u << 19) - 1;
        out[((size_t)b << 19) + lo] = 0.0f;
    }
}

// ---------------------------------------------------------------------------
extern "C" void kernel_launch(void* const* d_in, const int* in_sizes, int n_in,
                              void* d_out, int out_size, void* d_ws, size_t ws_size,
                              hipStream_t stream) {
    const float* x = (const float*)d_in[0];
    float* out = (float*)d_out;

    char* ws = (char*)d_ws;
    float* cdf      = (float*)ws;                                  // 134217728 B
    float* partial  = (float*)(ws + 134217728u);                   // 65536 B
    float* chunkOff = (float*)(ws + 134217728u + 65536u);          // 65536 B
    float* totals   = (float*)(ws + 134217728u + 2u * 65536u);     // 256 B

    k_chunksum <<<TOTAL_CHUNKS, 256, 0, stream>>>(x, partial);
    k_scanpart <<<B_ROWS,       256, 0, stream>>>(partial, chunkOff, totals);
    k_scan_wmma<<<TOTAL_CHUNKS, 256, 0, stream>>>(x, chunkOff, cdf);
    k_scale    <<<32768,        256, 0, stream>>>(x, totals, out);
    k_sample   <<<32768,        256, 0, stream>>>(cdf, totals, out);
}